// SocialCircleLayer_4320737100230
// MI455X (gfx1250) — compile-verified
//
#include <hip/hip_runtime.h>
#include <hip/hip_bf16.h>
#include <math.h>

typedef __attribute__((ext_vector_type(2))) float v2f;
typedef __attribute__((ext_vector_type(8))) float v8f;

#define PARTITIONS 8
#define NNEI 128
#define NT 8
#define NF 3
#define NC 128
#define TWO_PI_F 6.283185307179586f
#define EPS_COUNT 0.0001f

// ---------------------------------------------------------------------------
// Kernel 1: per-batch social-circle binning.
// grid = B blocks, block = 128 threads (one neighbor per thread, 4 waves).
// Reads 16 floats (64B) per neighbor via 4x b128 loads, reduces into LDS.
// ---------------------------------------------------------------------------
__global__ void __launch_bounds__(NNEI)
sc_bin_kernel(const float* __restrict__ nei, float* __restrict__ sc_out) {
    __shared__ float s[PARTITIONS][4];   // [vel, dist, dir, count]

    const int b = blockIdx.x;
    const int n = threadIdx.x;

    if (threadIdx.x < PARTITIONS * 4) {
        ((float*)s)[threadIdx.x] = 0.0f;
    }
    __syncthreads();

    const size_t base = ((size_t)b * NNEI + n) * (NT * 2);
    const float4* row = (const float4*)(nei + base);
    float4 q0 = row[0];   // t0.xy t1.xy
    float4 q1 = row[1];   // t2.xy t3.xy
    float4 q2 = row[2];   // t4.xy t5.xy
    float4 q3 = row[3];   // t6.xy t7.xy

    float total = q0.x + q0.y + q0.z + q0.w
                + q1.x + q1.y + q1.z + q1.w
                + q2.x + q2.y + q2.z + q2.w
                + q3.x + q3.y + q3.z + q3.w;
    bool active = (total != 0.0f);

    float p0x = q0.x, p0y = q0.y;
    float p7x = q3.z, p7y = q3.w;

    float vx = p7x - p0x, vy = p7y - p0y;
    float vel  = sqrtf(vx * vx + vy * vy);
    float dist = sqrtf(p7x * p7x + p7y * p7y);
    float dir  = atan2f(p7x, p7y);          // arctan2(pos[...,0], pos[...,1])
    if (dir < 0.0f) dir += TWO_PI_F;        // mod 2*pi

    int idx = (int)(dir / (TWO_PI_F / (float)PARTITIONS));

    if (active && idx >= 0 && idx < PARTITIONS) {
        atomicAdd(&s[idx][0], vel);
        atomicAdd(&s[idx][1], dist);
        atomicAdd(&s[idx][2], dir);
        atomicAdd(&s[idx][3], 1.0f);
    }
    __syncthreads();

    // social_circle[b][p][f] = sums[p][f] / (count[p] + eps)
    if (threadIdx.x < PARTITIONS * NF) {
        int p = threadIdx.x / NF;
        int f = threadIdx.x % NF;
        float inv = 1.0f / (s[p][3] + EPS_COUNT);
        sc_out[(size_t)b * (PARTITIONS * NF) + threadIdx.x] = s[p][f] * inv;
    }
}

// ---------------------------------------------------------------------------
// Kernel 2: f_sc = relu(social_circle @ W_ce + b_ce) via v_wmma_f32_16x16x4_f32
// Rows = B*8 (A is rows x 3, K padded to 4), cols = 128.
// block = 256 threads (8 waves); each wave: 16 rows, 8 WMMAs across N=128.
// All fragment loads are branchless (cndmask-selected offsets/values) so the
// hot loop has no exec-mask manipulation and loads stay clause-able.
// ---------------------------------------------------------------------------
__global__ void __launch_bounds__(256)
sc_gemm_kernel(const float* __restrict__ sc, const float* __restrict__ W,
               const float* __restrict__ bias, float* __restrict__ fsc,
               int totalRows) {
    const int lane   = threadIdx.x & 31;
    const int waveId = threadIdx.x >> 5;
    const int row0   = (blockIdx.x * 8 + waveId) * 16;

    const int half = lane >> 4;      // 0: lanes 0-15, 1: lanes 16-31
    const int l16  = lane & 15;

    if (row0 + 16 <= totalRows) {
        // ---- A fragment: 16x4 f32 (lanes 0-15: K=0,1; lanes 16-31: K=2,pad)
        const int arow = row0 + l16;
        const float f0 = sc[(size_t)arow * NF + 0];   // velocity
        const float f1 = sc[(size_t)arow * NF + 1];   // distance
        const float f2 = sc[(size_t)arow * NF + 2];   // direction
        v2f a;
        a.x = half ? f2 : f0;
        a.y = half ? 0.0f : f1;

        // branchless B-row offsets: half0 -> rows {0,1}; half1 -> rows {2,(pad)}
        const int brow_x = half ? 2 : 0;
        const int brow_y = half ? 2 : 1;   // half1 value discarded by select

        #pragma unroll
        for (int t = 0; t < NC / 16; ++t) {
            const int c = t * 16 + l16;

            // ---- B fragment: 4x16 f32 (VGPR r -> K=r / K=r+2 per half) ----
            const float wx = W[brow_x * NC + c];
            const float wy = W[brow_y * NC + c];
            v2f bf;
            bf.x = wx;
            bf.y = half ? 0.0f : wy;

            v8f acc = {};
            acc = __builtin_amdgcn_wmma_f32_16x16x4_f32(
                false, a, false, bf, (short)0, acc, false, false);

            // ---- epilogue: bias + relu, coalesced 64B row segments ----
            const float bv = bias[c];
            #pragma unroll
            for (int r = 0; r < 8; ++r) {
                const int orow = row0 + r + (half << 3);
                float v = acc[r] + bv;
                v = v > 0.0f ? v : 0.0f;
                fsc[(size_t)orow * NC + c] = v;
            }
        }
    } else if (row0 < totalRows) {
        // scalar tail (wave-uniform branch keeps EXEC=all-1s on the WMMA path)
        const int nrows = totalRows - row0;
        for (int i = lane; i < nrows * NC; i += 32) {
            const int r = row0 + i / NC;
            const int c = i % NC;
            float v = bias[c];
            #pragma unroll
            for (int k = 0; k < NF; ++k)
                v += sc[(size_t)r * NF + k] * W[k * NC + c];
            fsc[(size_t)r * NC + c] = v > 0.0f ? v : 0.0f;
        }
    }
}

// ---------------------------------------------------------------------------
// Launch: d_in = {trajs (unused), nei_trajs, W_ce, b_ce}
// d_out = [ f_sc (B*8*128) | social_circle (B*8*3) ]
// ---------------------------------------------------------------------------
extern "C" void kernel_launch(void* const* d_in, const int* in_sizes, int n_in,
                              void* d_out, int out_size, void* d_ws, size_t ws_size,
                              hipStream_t stream) {
    const float* nei  = (const float*)d_in[1];
    const float* W    = (const float*)d_in[2];
    const float* bias = (const float*)d_in[3];
    float* out = (float*)d_out;

    const int B = in_sizes[1] / (NNEI * NT * 2);

    float* fsc = out;                                   // B*8*128
    float* sc  = out + (size_t)B * PARTITIONS * NC;     // B*8*3

    sc_bin_kernel<<<B, NNEI, 0, stream>>>(nei, sc);

    const int totalRows = B * PARTITIONS;               // rows of A
    const int rowsPerBlock = 16 * 8;                    // 16 rows/wave * 8 waves
    const int nblocks = (totalRows + rowsPerBlock - 1) / rowsPerBlock;
    sc_gemm_kernel<<<nblocks, 256, 0, stream>>>(sc, W, bias, fsc, totalRows);
}